// SNU_53678501265715
// MI455X (gfx1250) — compile-verified
//
#include <hip/hip_runtime.h>

typedef float v2f __attribute__((ext_vector_type(2)));
typedef float v8f __attribute__((ext_vector_type(8)));
typedef unsigned int v4u __attribute__((ext_vector_type(4)));
typedef int v4i __attribute__((ext_vector_type(4)));
typedef int v8i __attribute__((ext_vector_type(8)));

#define B_DIM 64
#define I_DIM 512
#define T_DIM 512
#define H_DIM 1024
#define DECAYF 0.8f
#define HT 64          // H columns per workgroup (4 waves x 16)
#define TBLK 16        // t values per block (one WMMA M-tile)
#define NTHREADS 128   // 4 wave32s
#define KCHUNK 32      // K-steps per unrolled chunk (keeps W ds-offsets < 64KB)

// Issue a TDM 2D tile load: x[b, 0:512, t0:t0+16] (rows of 16 contiguous f32,
// row stride T floats) -> LDS row-major [i][t].  D# layout per CDNA5 ISA §8.3-8.5.
__device__ __forceinline__ void tdm_load_x_block(const float* gsrc, unsigned lds_byte_addr)
{
    const unsigned long long ga = (unsigned long long)gsrc;

    v4u g0;
    g0.x = 1u;                                   // count=1, user descriptor
    g0.y = lds_byte_addr;                        // lds_addr (bytes)
    g0.z = (unsigned)ga;                         // global_addr[31:0]
    g0.w = (unsigned)((ga >> 32) & 0x1FFFFFFu)   // global_addr[56:32]
         | (2u << 30);                           // type=2 ("image")

    v8i g1;
    g1[0] = (2 << 16);                           // wg_mask=0, data_size=2 (4B)
    g1[1] = (int)(512u << 16);                   // tensor_dim0[15:0]=512
    g1[2] = (int)(512u << 16);                   // dim0 hi=0 | tensor_dim1[15:0]=512
    g1[3] = (int)((unsigned)TBLK << 16);         // dim1 hi=0 | tile_dim0=16
    g1[4] = 512;                                 // tile_dim1=512 | tile_dim2=0
    g1[5] = 512;                                 // tensor_dim0_stride[31:0]=512 elems
    g1[6] = 0;                                   // stride0 hi | dim1_stride lo
    g1[7] = 0;

    v4i gz = {0, 0, 0, 0};                       // 2D tile: groups 2/3 unused

#if defined(__clang_major__) && (__clang_major__ >= 23)
    v8i gz8 = {0, 0, 0, 0, 0, 0, 0, 0};
    __builtin_amdgcn_tensor_load_to_lds(g0, g1, gz, gz, gz8, 0);
#else
    __builtin_amdgcn_tensor_load_to_lds(g0, g1, gz, gz, 0);
#endif
}

__launch_bounds__(NTHREADS)
__global__ void snu_fused_kernel(const float* __restrict__ x,
                                 const float* __restrict__ W,
                                 const float* __restrict__ bias,
                                 float* __restrict__ y_out)
{
    // W: K-pairs interleaved so B fragments are single ds_load_b64:
    //    element (i, c) lives at [ (i>>1)*HT + c ]*2 + (i&1)
    // X: row-major [i][t] (written by the TDM engine in its natural scan order)
    __shared__ float Wlds[I_DIM * HT];    // 128 KB: W[:, hbase:hbase+HT]
    __shared__ float Xlds[I_DIM * TBLK];  //  32 KB: x[b, :, t0:t0+16]
    // total 160 KB -> exactly 2 workgroups per 320 KB WGP

    const int tid   = threadIdx.x;
    const int lane  = tid & 31;
    const int wave  = tid >> 5;
    const int b     = blockIdx.y;
    const int hbase = blockIdx.x * HT;
    const int m     = lane & 15;           // M row (t) for A, N col (h) for B/D
    const int col   = wave * 16 + m;       // column within the HT tile
    const int khalf = lane >> 4;           // lanes 16-31 hold K+2,K+3

    // ---- stage W[:, hbase:hbase+HT] into LDS once (coalesced float4) ----
    for (int idx = tid; idx < (I_DIM * HT) / 4; idx += NTHREADS) {
        const int i  = idx >> 4;           // HT/4 = 16 float4 per row
        const int c4 = (idx & 15) << 2;
        const float4 wv = *(const float4*)(W + (size_t)i * H_DIM + hbase + c4);
        const int base = ((i >> 1) * HT + c4) * 2 + (i & 1);
        Wlds[base + 0] = wv.x;
        Wlds[base + 2] = wv.y;
        Wlds[base + 4] = wv.z;
        Wlds[base + 6] = wv.w;
    }

    const float bn = bias[hbase + col];
    float hs = 0.0f, ys = 0.0f;            // recurrent state for column `col`

    // per-lane fragment base pointers
    const float* const apb = Xlds + m + khalf * 32;        // A: +64 floats / k-step, pair +16
    const float* const wpb = Wlds + col * 2 + khalf * 128; // B: +256 floats / k-step

    const float* const xrow = x + (size_t)b * I_DIM * T_DIM;   // x[b,0,0]
    const unsigned xlds_addr = (unsigned)(unsigned long long)(&Xlds[0]);

    for (int tb = 0; tb < T_DIM / TBLK; ++tb) {
        const int t0 = tb * TBLK;

        __syncthreads();                   // W ready (tb==0) / Xlds readers done (tb>0)

        // ---- async-tensor DMA: x[b, :, t0:t0+16] -> Xlds, one instruction ----
        if (wave == 0) {
            tdm_load_x_block(xrow + t0, xlds_addr);
            __builtin_amdgcn_s_wait_tensorcnt(0);
        }
        __syncthreads();                   // Xlds visible to all waves

        // ---- prefetch next t-block of x into cache while we compute ----
        if (tb + 1 < T_DIM / TBLK) {
            const float* nx = xrow + t0 + TBLK;
            #pragma unroll
            for (int r = 0; r < I_DIM / NTHREADS; ++r) {
                __builtin_prefetch(nx + (size_t)(tid + r * NTHREADS) * T_DIM, 0, 1);
            }
        }

        // ---- D(16t x 16h) = sum over K=512 of A(16x4) * B(4x16), f32 WMMA ----
        // B fragments: ds_load_b64 (interleaved W), immediate offsets.
        // A fragments: two b32 loads 64B apart (row-major X) -> ds_load_2addr_b32.
        v8f acc = {0.f, 0.f, 0.f, 0.f, 0.f, 0.f, 0.f, 0.f};
        const float* ap = apb;
        const float* wp = wpb;
        #pragma unroll 1
        for (int cc = 0; cc < (I_DIM / 4) / KCHUNK; ++cc) {
            #pragma unroll
            for (int j = 0; j < KCHUNK; ++j) {
                v2f a;
                a.x = ap[j * 64];
                a.y = ap[j * 64 + 16];
                const v2f w = *(const v2f*)(wp + j * 256);
                acc = __builtin_amdgcn_wmma_f32_16x16x4_f32(
                          false, a, false, w, (short)0, acc, false, false);
            }
            ap += KCHUNK * 64;
            wp += KCHUNK * 256;
        }

        // ---- sequential scan over the 16 t's of this block, in registers ----
        // acc[r]: lanes 0-15 hold t=t0+r, lanes 16-31 hold t=t0+8+r
        #pragma unroll
        for (int r = 0; r < 8; ++r) {      // t = t0 + r (valid in low half)
            hs = fmaxf(acc[r] + DECAYF * hs * (1.0f - ys), 0.0f);
            const float z = hs + bn;
            ys = (z > 1.0f) ? z : 0.0f;
        }
        hs = __shfl(hs, m, 32);            // pass state low half -> all lanes
        ys = __shfl(ys, m, 32);
        #pragma unroll
        for (int r = 0; r < 8; ++r) {      // t = t0 + 8 + r (valid in high half)
            hs = fmaxf(acc[r] + DECAYF * hs * (1.0f - ys), 0.0f);
            const float z = hs + bn;
            ys = (z > 1.0f) ? z : 0.0f;
        }
        hs = __shfl(hs, m + 16, 32);       // pass state high half -> all lanes
        ys = __shfl(ys, m + 16, 32);
    }

    if (lane < 16) {
        y_out[(size_t)b * H_DIM + hbase + col] = ys;
    }
}

extern "C" void kernel_launch(void* const* d_in, const int* in_sizes, int n_in,
                              void* d_out, int out_size, void* d_ws, size_t ws_size,
                              hipStream_t stream) {
    (void)in_sizes; (void)n_in; (void)out_size; (void)d_ws; (void)ws_size;
    const float* x    = (const float*)d_in[0];
    const float* W    = (const float*)d_in[1];
    const float* bias = (const float*)d_in[2];
    float* y          = (float*)d_out;

    dim3 grid(H_DIM / HT, B_DIM);   // (16, 64) = 1024 workgroups
    dim3 block(NTHREADS);           // 4 wave32s
    hipLaunchKernelGGL(snu_fused_kernel, grid, block, 0, stream, x, W, bias, y);
}